// Net_56495999811606
// MI455X (gfx1250) — compile-verified
//
#include <hip/hip_runtime.h>

typedef __attribute__((ext_vector_type(2))) float v2f;
typedef __attribute__((ext_vector_type(8))) float v8f;

// ---------------- degree / norm precompute ----------------

__global__ void deg_init_kernel(float* dis, int n) {
  int i = blockIdx.x * blockDim.x + threadIdx.x;
  if (i < n) dis[i] = 1.0f;  // self-loop contributes 1 to every node's degree
}

__global__ void deg_count_kernel(const int* __restrict__ dst, float* dis, int e) {
  int i = blockIdx.x * blockDim.x + threadIdx.x;
  if (i < e) atomicAdd(&dis[dst[i]], 1.0f);
}

__global__ void deg_rsqrt_kernel(float* dis, int n) {
  int i = blockIdx.x * blockDim.x + threadIdx.x;
  if (i < n) dis[i] = rsqrtf(dis[i]);  // deg >= 1 always (self loops)
}

__global__ void norm_kernel(const int* __restrict__ src, const int* __restrict__ dst,
                            const float* __restrict__ dis, float* __restrict__ nrm, int e) {
  int i = blockIdx.x * blockDim.x + threadIdx.x;
  if (i < e) nrm[i] = dis[src[i]] * dis[dst[i]];
}

// ---------------- dense GEMM: out[N x CO] = A[N x CI] @ W[CI x CO] ----------------
// One wave computes one 16x16 tile via V_WMMA_F32_16X16X4_F32 (fp32, K=4/step).
// blockIdx.x selects the 16-row slab; wave id selects the 16-col slab.
// Requires N % 16 == 0 (N = 50000 = 3125 * 16) and EXEC all-ones (no divergence).
template <int CI, int CO>
__global__ void gemm_wmma_kernel(const float* __restrict__ A, const float* __restrict__ W,
                                 float* __restrict__ out) {
  const int lane = threadIdx.x & 31;
  const int wave = threadIdx.x >> 5;
  const int m0 = blockIdx.x << 4;
  const int n0 = wave << 4;
  const int half = lane >> 4;   // 0: lanes 0-15, 1: lanes 16-31
  const int l16 = lane & 15;

  // A frag: lane l16 = row m0+l16; v0/v1 hold K = k0 + half*2 + {0,1}
  const float* __restrict__ ap = A + (size_t)(m0 + l16) * CI + (half << 1);
  // B frag: lane l16 = col n0+l16; v0/v1 hold K = k0 + half*2 + {0,1}
  const float* __restrict__ wp = W + n0 + l16;

  v8f acc = {};
#pragma unroll 4
  for (int k0 = 0; k0 < CI; k0 += 4) {
    v2f a = *(const v2f*)(ap + k0);  // 8B-aligned: rows are CI-float aligned
    const int kb = k0 + (half << 1);
    v2f b;
    b.x = wp[(size_t)kb * CO];
    b.y = wp[(size_t)(kb + 1) * CO];
    acc = __builtin_amdgcn_wmma_f32_16x16x4_f32(false, a, false, b, (short)0, acc,
                                                false, false);
  }

  // D layout: vgpr v holds row m0 + v + half*8, col n0 + l16
  float* __restrict__ op = out + (size_t)(m0 + (half << 3)) * CO + n0 + l16;
#pragma unroll
  for (int v = 0; v < 8; ++v) op[(size_t)v * CO] = acc[v];
}

// ---------------- aggregation ----------------

// agg[i,c] = P[i,c] * dis[i]^2   (folds the self-loop term; also zero-fills)
__global__ void self_loop_init_kernel(const float* __restrict__ P,
                                      const float* __restrict__ dis,
                                      float* __restrict__ agg, int total, int logC) {
  int i = blockIdx.x * blockDim.x + threadIdx.x;
  if (i < total) {
    float d = dis[i >> logC];
    agg[i] = P[i] * d * d;
  }
}

// One wave per edge: coalesced 512B/256B gather of P[src], scaled atomic scatter
// into agg[dst]. src/dst/nrm reads are wave-uniform -> scalar path.
template <int C>
__global__ void scatter_edges_kernel(const float* __restrict__ P, const int* __restrict__ src,
                                     const int* __restrict__ dst, const float* __restrict__ nrm,
                                     float* agg, int e) {
  int t = blockIdx.x * blockDim.x + threadIdx.x;
  int wid = t >> 5;
  int lane = t & 31;
  if (wid >= e) return;
  int s = src[wid];
  int d = dst[wid];
  float w = nrm[wid];
  if constexpr (C == 128) {
    float4 v = *((const float4*)(P + (size_t)s * C) + lane);
    float* o = agg + (size_t)d * C + (lane << 2);
    atomicAdd(o + 0, v.x * w);
    atomicAdd(o + 1, v.y * w);
    atomicAdd(o + 2, v.z * w);
    atomicAdd(o + 3, v.w * w);
  } else {  // C == 64
    float2 v = *((const float2*)(P + (size_t)s * C) + lane);
    float* o = agg + (size_t)d * C + (lane << 1);
    atomicAdd(o + 0, v.x * w);
    atomicAdd(o + 1, v.y * w);
  }
}

// h = leaky_relu(h + b)  (or just +b when relu==0); C is a power of two
__global__ void bias_act_kernel(float* h, const float* __restrict__ b, int total, int cmask,
                                int relu) {
  int i = blockIdx.x * blockDim.x + threadIdx.x;
  if (i < total) {
    float v = h[i] + b[i & cmask];
    h[i] = (relu && v < 0.0f) ? 0.2f * v : v;
  }
}

// ---------------- decode: logits[j] = dot64(z[e0[j]], z[e1[j]]) ----------------
// One wave per query edge; lane holds 2 channels; wave32 shfl_xor reduction.
__global__ void decode_kernel(const float* __restrict__ z, const int* __restrict__ pos,
                              const int* __restrict__ neg, float* __restrict__ out, int el) {
  int t = blockIdx.x * blockDim.x + threadIdx.x;
  int wid = t >> 5;
  int lane = t & 31;
  int total = 2 * el;
  if (wid >= total) return;
  int a, b;
  if (wid < el) {
    a = pos[wid];
    b = pos[el + wid];
  } else {
    int j = wid - el;
    a = neg[j];
    b = neg[el + j];
  }
  float2 za = *((const float2*)(z + (size_t)a * 64) + lane);
  float2 zb = *((const float2*)(z + (size_t)b * 64) + lane);
  float p = za.x * zb.x + za.y * zb.y;
#pragma unroll
  for (int off = 16; off > 0; off >>= 1) p += __shfl_xor(p, off, 32);
  if (lane == 0) out[wid] = p;
}

// ---------------- driver ----------------

extern "C" void kernel_launch(void* const* d_in, const int* in_sizes, int n_in,
                              void* d_out, int out_size, void* d_ws, size_t ws_size,
                              hipStream_t stream) {
  const float* x  = (const float*)d_in[0];
  const int*   ei = (const int*)d_in[1];
  const int*   pe = (const int*)d_in[2];
  const int*   ne = (const int*)d_in[3];
  const float* W1 = (const float*)d_in[4];
  const float* b1 = (const float*)d_in[5];
  const float* W  = (const float*)d_in[6];
  const float* b  = (const float*)d_in[7];
  const float* W2 = (const float*)d_in[8];
  const float* b2 = (const float*)d_in[9];

  const int N  = in_sizes[0] / 128;
  const int E  = in_sizes[1] / 2;
  const int EL = in_sizes[2] / 2;
  const int* src = ei;
  const int* dst = ei + E;

  size_t off = 0;
  char* base = (char*)d_ws;
  auto carve = [&](size_t bytes) -> float* {
    float* p = (float*)(base + off);
    off += (bytes + 255) & ~(size_t)255;
    return p;
  };
  float* dis = carve((size_t)N * 4);
  float* nrm = carve((size_t)E * 4);
  float* P   = carve((size_t)N * 128 * 4);  // GEMM output
  float* Q   = carve((size_t)N * 128 * 4);  // ping
  float* R   = carve((size_t)N * 128 * 4);  // pong
  (void)ws_size; (void)n_in; (void)out_size;

  const int BT = 256;
  const int nb_N   = (N + BT - 1) / BT;
  const int nb_E   = (E + BT - 1) / BT;
  const int nb_NC  = (N * 128 + BT - 1) / BT;
  const int nb_NC2 = (N * 64 + BT - 1) / BT;
  const int nb_Ew  = (E * 32 + BT - 1) / BT;       // wave per edge
  const int nb_Dw  = (2 * EL * 32 + BT - 1) / BT;  // wave per query edge
  const int mtiles = N / 16;                        // 3125

  // symmetric normalization
  deg_init_kernel<<<nb_N, BT, 0, stream>>>(dis, N);
  deg_count_kernel<<<nb_E, BT, 0, stream>>>(dst, dis, E);
  deg_rsqrt_kernel<<<nb_N, BT, 0, stream>>>(dis, N);
  norm_kernel<<<nb_E, BT, 0, stream>>>(src, dst, dis, nrm, E);

  // layer 1: x @ W1 -> aggregate -> +b1, leaky_relu  => Q
  gemm_wmma_kernel<128, 128><<<mtiles, 256, 0, stream>>>(x, W1, P);
  self_loop_init_kernel<<<nb_NC, BT, 0, stream>>>(P, dis, Q, N * 128, 7);
  scatter_edges_kernel<128><<<nb_Ew, BT, 0, stream>>>(P, src, dst, nrm, Q, E);
  bias_act_kernel<<<nb_NC, BT, 0, stream>>>(Q, b1, N * 128, 127, 1);

  // layer 2: Q @ W -> aggregate -> +b, leaky_relu  => R
  gemm_wmma_kernel<128, 128><<<mtiles, 256, 0, stream>>>(Q, W, P);
  self_loop_init_kernel<<<nb_NC, BT, 0, stream>>>(P, dis, R, N * 128, 7);
  scatter_edges_kernel<128><<<nb_Ew, BT, 0, stream>>>(P, src, dst, nrm, R, E);
  bias_act_kernel<<<nb_NC, BT, 0, stream>>>(R, b, N * 128, 127, 1);

  // layer 3: R @ W -> aggregate -> +b, leaky_relu  => Q
  gemm_wmma_kernel<128, 128><<<mtiles, 256, 0, stream>>>(R, W, P);
  self_loop_init_kernel<<<nb_NC, BT, 0, stream>>>(P, dis, Q, N * 128, 7);
  scatter_edges_kernel<128><<<nb_Ew, BT, 0, stream>>>(P, src, dst, nrm, Q, E);
  bias_act_kernel<<<nb_NC, BT, 0, stream>>>(Q, b, N * 128, 127, 1);

  // layer 4 (output, no activation): Q @ W2 -> aggregate -> +b2  => R[:, :64]
  gemm_wmma_kernel<128, 64><<<mtiles, 128, 0, stream>>>(Q, W2, P);
  self_loop_init_kernel<<<nb_NC2, BT, 0, stream>>>(P, dis, R, N * 64, 6);
  scatter_edges_kernel<64><<<nb_Ew, BT, 0, stream>>>(P, src, dst, nrm, R, E);
  bias_act_kernel<<<nb_NC2, BT, 0, stream>>>(R, b2, N * 64, 63, 0);

  // decode
  decode_kernel<<<nb_Dw, BT, 0, stream>>>(R, pe, ne, (float*)d_out, EL);
}